// ODEFunc_5488968204447
// MI455X (gfx1250) — compile-verified
//
#include <hip/hip_runtime.h>

// MI455X / gfx1250, wave32. FP32 WMMA (V_WMMA_F32_16X16X4_F32) for all matmul
// layers, VALU transcendentals, wave-private LDS transposes (no barriers).

typedef float v2f __attribute__((ext_vector_type(2)));
typedef float v8f __attribute__((ext_vector_type(8)));

// 8-arg form per probe: (neg_a, A, neg_b, B, c_mod, C, reuse_a, reuse_b)
#define WMMA_F32(a, b, c) \
  __builtin_amdgcn_wmma_f32_16x16x4_f32(false, (a), false, (b), (short)0, (c), false, false)

__device__ __forceinline__ float fast_rcp(float x) { return __builtin_amdgcn_rcpf(x); }
__device__ __forceinline__ float elu_f(float x) { return x > 0.f ? x : (__expf(x) - 1.f); }
__device__ __forceinline__ float tanh_f(float x) {
  float ax = __builtin_fabsf(x);
  float e  = __expf(-2.f * ax);
  float t  = (1.f - e) * fast_rcp(1.f + e);
  return x < 0.f ? -t : t;
}
__device__ __forceinline__ float sigmoid_f(float x) {
  return fast_rcp(1.f + __expf(-x));
}

#define WAVES_PER_BLOCK 8
#define LDS_STRIDE 20            // 16 feature slots + 4 spare (bmat stash), bank-conflict-free
#define LDS_PER_WAVE 352         // 16*20 rows + 32 floats for the g stash

__global__ __launch_bounds__(256) void odefunc_kernel(
    const float* __restrict__ x,
    const float* __restrict__ mW0, const float* __restrict__ mb0,
    const float* __restrict__ mW1, const float* __restrict__ mb1,
    const float* __restrict__ mW2, const float* __restrict__ mb2,
    const float* __restrict__ pW1, const float* __restrict__ pb1,
    const float* __restrict__ pW2, const float* __restrict__ pb2,
    const float* __restrict__ pW3, const float* __restrict__ pb3,
    float* __restrict__ out, int npts)
{
  __shared__ float ldsbuf[WAVES_PER_BLOCK * LDS_PER_WAVE];
  const int lane = threadIdx.x & 31;
  const int wid  = threadIdx.x >> 5;
  const int half = lane >> 4;        // 0: lanes 0-15, 1: lanes 16-31
  const int col  = lane & 15;        // WMMA column (N) held by this lane
  float* T = &ldsbuf[wid * LDS_PER_WAVE];

  // ---------- hoisted wave-invariant WMMA B fragments ----------
  // Fragment element at (vgpr v, this lane): K = kbase + v + 2*half, N = col.
  auto mkB = [&](int g, auto f) -> v2f {
    v2f b; int k0 = 4 * g + 2 * half;
    b.x = f(k0 + 0);
    b.y = f(k0 + 1);
    return b;
  };

  // L0: W0^T (2x16, zero-padded K rows 2,3); cols >=8 zero.
  v2f bL0 = mkB(0, [&](int k) { return (k < 2 && col < 8) ? mW0[col * 2 + k] : 0.f; });
  // L1: W1^T (8x16) split into two K-groups.
  v2f bL1[2]; for (int g = 0; g < 2; ++g)
    bL1[g] = mkB(g, [&](int k) { return (col < 8) ? mW1[col * 8 + k] : 0.f; });
  // L2: W2^T (8x4).
  v2f bL2[2]; for (int g = 0; g < 2; ++g)
    bL2[g] = mkB(g, [&](int k) { return (col < 4) ? mW2[col * 8 + k] : 0.f; });
  // P0: both variants batched: cols 0-7 = pW1[0]^T, cols 8-15 = pW1[1]^T.
  v2f bP0 = mkB(0, [&](int k) {
    return (k < 2) ? pW1[(col >> 3) * 16 + (col & 7) * 2 + k] : 0.f; });
  // P1 forward: block-diag(pW2[0]^T, pW2[1]^T), K=16 -> 4 groups.
  v2f bP1[4]; for (int g = 0; g < 4; ++g)
    bP1[g] = mkB(g, [&](int k) {
      int kv = k >> 3, nv = col >> 3;
      return (kv == nv) ? pW2[kv * 64 + (col & 7) * 8 + (k & 7)] : 0.f; });
  // Backward 1: block-diag(pW2[0], pW2[1]) (not transposed).
  v2f bB1[4]; for (int g = 0; g < 4; ++g)
    bB1[g] = mkB(g, [&](int k) {
      int kv = k >> 3, nv = col >> 3;
      return (kv == nv) ? pW2[kv * 64 + (k & 7) * 8 + (col & 7)] : 0.f; });
  // Backward 2: rows p'=8v+p -> pW1[v][p][i] in cols 0,1; both variants sum into
  // the same output columns (handles the mean over v up to a factor 0.5).
  v2f bB2[4]; for (int g = 0; g < 4; ++g)
    bB2[g] = mkB(g, [&](int k) {
      return (col < 2) ? pW1[(k >> 3) * 16 + (k & 7) * 2 + col] : 0.f; });

  // Bias values broadcast along rows (one scalar per lane -> all 8 C VGPRs).
  const float cm0 = (col < 8) ? mb0[col] : 0.f;
  const float cm1 = (col < 8) ? mb1[col] : 0.f;
  const float cm2 = (col < 4) ? mb2[col] : 0.f;
  const float cp1 = pb1[col];                 // (NV,HP) flat == 16
  const float cp2 = pb2[col];                 // (NV,1,HP) flat == 16
  // Per-lane head weights: lane = (point=col, variant=half).
  float w3[8];
  for (int j = 0; j < 8; ++j) w3[j] = pW3[half * 8 + j];
  const float pb3v = pb3[half];

  auto storeD = [&](const v8f& dd) {          // D layout -> LDS, one row per point, one column per feature
    for (int r = 0; r < 8; ++r) T[(r + 8 * half) * LDS_STRIDE + col] = dd[r];
  };
  auto loadA = [&](int g) -> v2f {            // LDS -> A fragment for K-group g
    v2f a; int f = 4 * g + 2 * half;
    a.x = T[col * LDS_STRIDE + f + 0];
    a.y = T[col * LDS_STRIDE + f + 1];
    return a;
  };
  auto biasC = [&](float b) -> v8f {
    v8f c; for (int r = 0; r < 8; ++r) c[r] = b; return c;
  };

  const int ntiles = (npts + 15) >> 4;
  const int wstride = gridDim.x * WAVES_PER_BLOCK;
  for (int tile = blockIdx.x * WAVES_PER_BLOCK + wid; tile < ntiles; tile += wstride) {
    const int base = tile << 4;
    const int pidx = min(base + col, npts - 1);

    // A fragment for x: K=0,1 valid (lanes 0-15), K=2,3 zero (lanes 16-31).
    v2f ax;
    {
      float2 p = ((const float2*)x)[pidx];
      ax.x = half ? 0.f : p.x;
      ax.y = half ? 0.f : p.y;
    }

    // ---- main net ----
    v8f d = WMMA_F32(ax, bL0, biasC(cm0));                   // 2 -> 8
    for (int r = 0; r < 8; ++r) d[r] = elu_f(d[r]);
    storeD(d);
    d = biasC(cm1);                                          // 8 -> 8
    d = WMMA_F32(loadA(0), bL1[0], d);
    d = WMMA_F32(loadA(1), bL1[1], d);
    for (int r = 0; r < 8; ++r) d[r] = elu_f(d[r]);
    storeD(d);
    d = biasC(cm2);                                          // 8 -> 4 (bmat)
    d = WMMA_F32(loadA(0), bL2[0], d);
    d = WMMA_F32(loadA(1), bL2[1], d);
    if (col < 4)                                             // stash b1..b4
      for (int r = 0; r < 8; ++r) T[(r + 8 * half) * LDS_STRIDE + 16 + col] = d[r];

    // ---- potential nets, forward (both variants in cols 0-7 / 8-15) ----
    v8f f1 = WMMA_F32(ax, bP0, biasC(cp1));
    for (int r = 0; r < 8; ++r) f1[r] = tanh_f(f1[r]);
    storeD(f1);
    v8f f2 = biasC(cp2);
    for (int g = 0; g < 4; ++g) f2 = WMMA_F32(loadA(g), bP1[g], f2);
    for (int r = 0; r < 8; ++r) f2[r] = tanh_f(f2[r]);
    storeD(f2);

    // ---- scalar head per (point, variant) lane; builds backward seed G ----
    float fr[8];
    for (int j = 0; j < 8; ++j) fr[j] = T[col * LDS_STRIDE + half * 8 + j];
    float dot = pb3v;
    for (int j = 0; j < 8; ++j) dot += fr[j] * w3[j];
    float f3 = sigmoid_f(dot);
    float s  = sigmoid_f(f3);
    float ss = s * (1.f - s);
    for (int j = 0; j < 8; ++j)
      T[col * LDS_STRIDE + half * 8 + j] = ss * w3[j] * (1.f - fr[j] * fr[j]);

    // ---- backward ----
    v8f gb = biasC(0.f);
    for (int g = 0; g < 4; ++g) gb = WMMA_F32(loadA(g), bB1[g], gb);
    for (int r = 0; r < 8; ++r) gb[r] = gb[r] * (1.f - f1[r] * f1[r]);
    storeD(gb);
    v8f g5 = biasC(0.f);
    for (int g = 0; g < 4; ++g) g5 = WMMA_F32(loadA(g), bB2[g], g5);
    if (col < 2)                                             // stash 2g
      for (int r = 0; r < 8; ++r) T[320 + (r + 8 * half) * 2 + col] = g5[r];

    // ---- final combine: lane = (point=col, component=half) ----
    float b1 = T[col * LDS_STRIDE + 16];
    float b2 = T[col * LDS_STRIDE + 17];
    float b3 = T[col * LDS_STRIDE + 18];
    float b4 = T[col * LDS_STRIDE + 19];
    float g0 = 0.5f * T[320 + col * 2 + 0];
    float g1 = 0.5f * T[320 + col * 2 + 1];
    float m2 = b1 * b2 + b3 * b4;
    float o  = half ? (m2 * g0 + (b2 * b2 + b4 * b4) * g1)
                    : ((b1 * b1 + b3 * b3) * g0 + m2 * g1);
    if (base + col < npts) out[(size_t)(base + col) * 2 + half] = o;
  }
}

extern "C" void kernel_launch(void* const* d_in, const int* in_sizes, int n_in,
                              void* d_out, int out_size, void* d_ws, size_t ws_size,
                              hipStream_t stream) {
  (void)n_in; (void)out_size; (void)d_ws; (void)ws_size;
  const float* x   = (const float*)d_in[1];
  const float* mW0 = (const float*)d_in[2];
  const float* mb0 = (const float*)d_in[3];
  const float* mW1 = (const float*)d_in[4];
  const float* mb1 = (const float*)d_in[5];
  const float* mW2 = (const float*)d_in[6];
  const float* mb2 = (const float*)d_in[7];
  const float* pW1 = (const float*)d_in[8];
  const float* pb1 = (const float*)d_in[9];
  const float* pW2 = (const float*)d_in[10];
  const float* pb2 = (const float*)d_in[11];
  const float* pW3 = (const float*)d_in[12];
  const float* pb3 = (const float*)d_in[13];

  int npts   = in_sizes[1] / 2;                 // x is (B, 2)
  int ntiles = (npts + 15) / 16;
  int blocks = (ntiles + WAVES_PER_BLOCK - 1) / WAVES_PER_BLOCK;
  if (blocks > 2048) blocks = 2048;             // grid-stride over 16-point tiles
  if (blocks < 1) blocks = 1;

  odefunc_kernel<<<blocks, 256, 0, stream>>>(
      x, mW0, mb0, mW1, mb1, mW2, mb2,
      pW1, pb1, pW2, pb2, pW3, pb3,
      (float*)d_out, npts);
}